// HeteroRegressor_4088808866423
// MI455X (gfx1250) — compile-verified
//
#include <hip/hip_runtime.h>
#include <hip/hip_bf16.h>

#define DD 128

typedef float v2f __attribute__((ext_vector_type(2)));
typedef float v8f __attribute__((ext_vector_type(8)));

// ---------------------------------------------------------------------------
// Degree accumulation: one thread per edge, HW float atomics.
// ---------------------------------------------------------------------------
__global__ void deg_kernel(const int* __restrict__ src, const int* __restrict__ dst,
                           float* __restrict__ degS, float* __restrict__ degD, int E) {
    int t = blockIdx.x * blockDim.x + threadIdx.x;
    if (t < E) {
        unsafeAtomicAdd(&degS[src[t]], 1.0f);
        unsafeAtomicAdd(&degD[dst[t]], 1.0f);
    }
}

// deg -> 1/sqrt(max(deg,1)) in place (runs over all 8 norm arrays at once)
__global__ void norm_kernel(float* __restrict__ d, int n) {
    int t = blockIdx.x * blockDim.x + threadIdx.x;
    if (t < n) d[t] = rsqrtf(fmaxf(d[t], 1.0f));
}

// ---------------------------------------------------------------------------
// Edge scatter: one wave32 per edge. Lane loads float4 of the source row
// (32 lanes * 4 floats = 128 = D), scales by norm_src, atomically adds into
// the destination row of the aggregation buffer.
// ---------------------------------------------------------------------------
__global__ void scatter_kernel(const float* __restrict__ h, const float* __restrict__ ns,
                               const int* __restrict__ src, const int* __restrict__ dst,
                               float* __restrict__ agg, int E) {
    int t = blockIdx.x * blockDim.x + threadIdx.x;
    int e = t >> 5;
    int lane = t & 31;
    if (e >= E) return;
    int s = src[e];
    int d = dst[e];
    float sc = ns[s];
    const float4* hv = (const float4*)(h + (size_t)s * DD);
    float4 v = hv[lane];
    float* ag = agg + (size_t)d * DD + lane * 4;
    unsafeAtomicAdd(ag + 0, v.x * sc);
    unsafeAtomicAdd(ag + 1, v.y * sc);
    unsafeAtomicAdd(ag + 2, v.z * sc);
    unsafeAtomicAdd(ag + 3, v.w * sc);
}

// ---------------------------------------------------------------------------
// Fused dual-relation GEMM via V_WMMA_F32_16X16X4_F32:
//   out = maybe_relu( (aggP * ndP) @ WP + bP + (aggQ * ndQ) @ WQ + bQ )
// Block = 256 threads (8 waves). Block owns one 16-row tile; wave w owns the
// 16-column tile starting at w*16. A tiles staged in LDS (stride 129 padding
// to avoid 16-way bank conflicts on A-fragment reads).
//
// f32 WMMA operand layout (ISA 7.12.2):
//   A 16x4  : lanes 0-15 row M=lane {K0,K1}; lanes 16-31 row M=lane-16 {K2,K3}
//   B 4x16  : mirrored (VGPR j, half h -> K = 2h+j, N = lane&15)
//   C/D 16x16: VGPR v -> row v (lanes 0-15) / row v+8 (lanes 16-31), N = lane&15
// ---------------------------------------------------------------------------
__global__ __launch_bounds__(256)
void gemm2_kernel(const float* __restrict__ aggP, const float* __restrict__ ndP,
                  const float* __restrict__ WP,   const float* __restrict__ bP,
                  const float* __restrict__ aggQ, const float* __restrict__ ndQ,
                  const float* __restrict__ WQ,   const float* __restrict__ bQ,
                  float* __restrict__ out, int nrows, int relu) {
    __shared__ float sP[16 * 129];
    __shared__ float sQ[16 * 129];
    int row0 = blockIdx.x * 16;
    int tid = threadIdx.x;

    // Cooperative load of both 16x128 A tiles, scaled by dst norms.
    for (int i = tid; i < 16 * DD; i += 256) {
        int r = i >> 7;           // 0..15
        int k = i & (DD - 1);     // 0..127
        int gr = row0 + r;
        if (gr < nrows) {
            sP[r * 129 + k] = aggP[(size_t)gr * DD + k] * ndP[gr];
            sQ[r * 129 + k] = aggQ[(size_t)gr * DD + k] * ndQ[gr];
        } else {
            sP[r * 129 + k] = 0.0f;
            sQ[r * 129 + k] = 0.0f;
        }
    }
    __syncthreads();

    int wave = tid >> 5;
    int lane = tid & 31;
    int nb = wave * 16;       // output column base for this wave
    int m = lane & 15;
    int h = lane >> 4;

    v8f acc = {0.f, 0.f, 0.f, 0.f, 0.f, 0.f, 0.f, 0.f};

#pragma unroll 4
    for (int kk = 0; kk < DD; kk += 4) {
        int k0 = kk + 2 * h;
        v2f aP, aQ, bPf, bQf;
        aP.x = sP[m * 129 + k0];
        aP.y = sP[m * 129 + k0 + 1];
        aQ.x = sQ[m * 129 + k0];
        aQ.y = sQ[m * 129 + k0 + 1];
        bPf.x = WP[(size_t)(k0    ) * DD + nb + m];
        bPf.y = WP[(size_t)(k0 + 1) * DD + nb + m];
        bQf.x = WQ[(size_t)(k0    ) * DD + nb + m];
        bQf.y = WQ[(size_t)(k0 + 1) * DD + nb + m];
        acc = __builtin_amdgcn_wmma_f32_16x16x4_f32(false, aP, false, bPf,
                                                    (short)0, acc, false, false);
        acc = __builtin_amdgcn_wmma_f32_16x16x4_f32(false, aQ, false, bQf,
                                                    (short)0, acc, false, false);
    }

    int n = nb + m;
    float bias = bP[n] + bQ[n];
#pragma unroll
    for (int v = 0; v < 8; ++v) {
        float x = acc[v] + bias;
        if (relu) x = fmaxf(x, 0.0f);
        int row = row0 + v + 8 * h;
        if (row < nrows) out[(size_t)row * DD + n] = x;
    }
}

// ---------------------------------------------------------------------------
// Column sums of hA + hB (128 threads = 128 columns; coalesced row sweeps).
// ---------------------------------------------------------------------------
__global__ void colsum_kernel(const float* __restrict__ hA, const float* __restrict__ hB,
                              float* __restrict__ colsum, int nrows) {
    int col = threadIdx.x;          // blockDim.x == 128
    int r0 = blockIdx.x * 256;
    int rend = min(r0 + 256, nrows);
    float s = 0.0f;
    for (int r = r0; r < rend; ++r)
        s += hA[(size_t)r * DD + col] + hB[(size_t)r * DD + col];
    unsafeAtomicAdd(&colsum[col], s);
}

__global__ void final_kernel(const float* __restrict__ colsum, const float* __restrict__ Wr,
                             const float* __restrict__ br, float* __restrict__ out, float invN) {
    __shared__ float red[128];
    int t = threadIdx.x;
    red[t] = colsum[t] * invN * Wr[t];
    __syncthreads();
    for (int s = 64; s > 0; s >>= 1) {
        if (t < s) red[t] += red[t + s];
        __syncthreads();
    }
    if (t == 0) out[0] = red[0] + br[0];
}

// ---------------------------------------------------------------------------
extern "C" void kernel_launch(void* const* d_in, const int* in_sizes, int n_in,
                              void* d_out, int out_size, void* d_ws, size_t ws_size,
                              hipStream_t stream) {
    const float* featA = (const float*)d_in[0];
    const float* featB = (const float*)d_in[1];
    const int* srcR[4] = {(const int*)d_in[2], (const int*)d_in[4],
                          (const int*)d_in[6], (const int*)d_in[8]};
    const int* dstR[4] = {(const int*)d_in[3], (const int*)d_in[5],
                          (const int*)d_in[7], (const int*)d_in[9]};
    const float* W  = (const float*)d_in[10];
    const float* b  = (const float*)d_in[11];
    const float* Wr = (const float*)d_in[12];
    const float* br = (const float*)d_in[13];

    const int Nn = in_sizes[0] / DD;
    const int E  = in_sizes[2];
    const int L  = in_sizes[10] / (4 * DD * DD);

    // ---- carve workspace ----
    float* ws = (float*)d_ws;
    size_t off = 0;
    float* ns[4];
    float* nd[4];
    for (int r = 0; r < 4; ++r) {
        ns[r] = ws + off; off += (size_t)Nn;
        nd[r] = ws + off; off += (size_t)Nn;
    }
    float* norms_base = ws;                 // 8*Nn contiguous
    float* agg0 = ws + off; off += (size_t)Nn * DD;
    float* agg1 = ws + off; off += (size_t)Nn * DD;
    float* hA0  = ws + off; off += (size_t)Nn * DD;
    float* hB0  = ws + off; off += (size_t)Nn * DD;
    float* hA1  = ws + off; off += (size_t)Nn * DD;
    float* hB1  = ws + off; off += (size_t)Nn * DD;
    float* colsum = ws + off; off += DD;

    const size_t featBytes = (size_t)Nn * DD * sizeof(float);

    // ---- degrees -> norms ----
    hipMemsetAsync(norms_base, 0, 8 * (size_t)Nn * sizeof(float), stream);
    int eb = (E + 255) / 256;
    for (int r = 0; r < 4; ++r)
        deg_kernel<<<eb, 256, 0, stream>>>(srcR[r], dstR[r], ns[r], nd[r], E);
    norm_kernel<<<(8 * Nn + 255) / 256, 256, 0, stream>>>(norms_base, 8 * Nn);

    // ---- layers ----
    const float* hA = featA;
    const float* hB = featB;
    float* outA = hA0; float* outB = hB0;
    float* altA = hA1; float* altB = hB1;

    const int sblocks = (int)(((size_t)E * 32 + 255) / 256);
    const int gblocks = (Nn + 15) / 16;

    for (int l = 0; l < L; ++l) {
        const int relu = (l < L - 1) ? 1 : 0;
        const float* Wl = W + (size_t)l * 4 * DD * DD;
        const float* bl = b + (size_t)l * 4 * DD;

        // newA = conv(hB via r1) + conv(hA via r2)
        hipMemsetAsync(agg0, 0, featBytes, stream);
        hipMemsetAsync(agg1, 0, featBytes, stream);
        scatter_kernel<<<sblocks, 256, 0, stream>>>(hB, ns[1], srcR[1], dstR[1], agg0, E);
        scatter_kernel<<<sblocks, 256, 0, stream>>>(hA, ns[2], srcR[2], dstR[2], agg1, E);
        gemm2_kernel<<<gblocks, 256, 0, stream>>>(
            agg0, nd[1], Wl + 1 * DD * DD, bl + 1 * DD,
            agg1, nd[2], Wl + 2 * DD * DD, bl + 2 * DD,
            outA, Nn, relu);

        // newB = conv(hA via r0) + conv(hB via r3)
        hipMemsetAsync(agg0, 0, featBytes, stream);
        hipMemsetAsync(agg1, 0, featBytes, stream);
        scatter_kernel<<<sblocks, 256, 0, stream>>>(hA, ns[0], srcR[0], dstR[0], agg0, E);
        scatter_kernel<<<sblocks, 256, 0, stream>>>(hB, ns[3], srcR[3], dstR[3], agg1, E);
        gemm2_kernel<<<gblocks, 256, 0, stream>>>(
            agg0, nd[0], Wl + 0 * DD * DD, bl + 0 * DD,
            agg1, nd[3], Wl + 3 * DD * DD, bl + 3 * DD,
            outB, Nn, relu);

        hA = outA; hB = outB;
        float* tA = outA; outA = altA; altA = tA;
        float* tB = outB; outB = altB; altB = tB;
    }

    // ---- readout: (mean(hA) + mean(hB)) @ Wr + br ----
    hipMemsetAsync(colsum, 0, DD * sizeof(float), stream);
    colsum_kernel<<<(Nn + 255) / 256, 128, 0, stream>>>(hA, hB, colsum, Nn);
    final_kernel<<<1, 128, 0, stream>>>(colsum, Wr, br, (float*)d_out, 1.0f / (float)Nn);
}